// GNN_GCNConv_homogen_72971494359491
// MI455X (gfx1250) — compile-verified
//
#include <hip/hip_runtime.h>

#define N_NODES 100000
#define N_EDGES 1600000
#define F_IN 256
#define F_HID 128
#define F_OUT 64

typedef float v2f __attribute__((ext_vector_type(2)));
typedef float v8f __attribute__((ext_vector_type(8)));
typedef unsigned int u32x4 __attribute__((ext_vector_type(4)));
typedef int i32x4 __attribute__((ext_vector_type(4)));
typedef int i32x8 __attribute__((ext_vector_type(8)));

// Use the Tensor Data Mover for weight-tile staging when the builtin exists.
// This toolchain (clang-23 / therock-10.0) exposes the 6-arg form:
//   (uint32x4 g0, int32x8 g1, int32x4 g2, int32x4 g3, int32x8, i32 cpol)
#ifndef USE_TDM
#if defined(__has_builtin)
#if __has_builtin(__builtin_amdgcn_tensor_load_to_lds) && __has_builtin(__builtin_amdgcn_s_wait_tensorcnt)
#define USE_TDM 1
#else
#define USE_TDM 0
#endif
#else
#define USE_TDM 0
#endif
#endif

// ---------------- utility kernels ----------------

__global__ __launch_bounds__(256) void fill_kernel(float* __restrict__ p, float v, long long n) {
    long long i = (long long)blockIdx.x * blockDim.x + threadIdx.x;
    if (i < n) p[i] = v;
}

__global__ __launch_bounds__(256) void deg_accum_kernel(const int* __restrict__ dst,
                                                        float* __restrict__ deg, int E) {
    int e = blockIdx.x * blockDim.x + threadIdx.x;
    if (e < E) atomicAdd(&deg[dst[e]], 1.0f);
}

__global__ __launch_bounds__(256) void dinv_kernel(float* __restrict__ deg, int n) {
    int i = blockIdx.x * blockDim.x + threadIdx.x;
    if (i < n) deg[i] = rsqrtf(deg[i]);   // deg >= 1 (self loop), matches reference
}

// ---------------- fp32 WMMA GEMM: C[M,Nc] = A[M,K] @ W[K,Nc] (+bias) ----------------
// Block = 8 waves; all waves share one 16-col tile of W, staged once into LDS by the
// Tensor Data Mover (D# 2D tile: 16 x K, stride Nc), then each wave computes a 16x16
// C tile via V_WMMA_F32_16X16X4_F32, K stepped by 4.
// A frag (16x4, 2 VGPR): lanes 0-15 -> row M=lane, K={k,k+1}; lanes 16-31 -> K={k+2,k+3}.
// B frag (4x16, 2 VGPR): lanes 0-15 -> col N=lane, K={k,k+1}; lanes 16-31 -> K={k+2,k+3}.
// C/D (8 VGPR): VGPR j -> M = j (lanes 0-15) / j+8 (lanes 16-31), N = lane&15.
__global__ __launch_bounds__(256)
void gemm_wmma_f32(const float* __restrict__ A, const float* __restrict__ W,
                   const float* __restrict__ bias, float* __restrict__ C,
                   int M, int K, int Nc) {
    __shared__ float Bs[256 * 16];            // staged W tile, [k][n], K <= 256 (16 KB)

    const int wave = threadIdx.x >> 5;
    const int lane = threadIdx.x & 31;
    const int col0 = blockIdx.y * 16;

    // ---- stage B tile (K rows x 16 cols of W) into LDS ----
#if USE_TDM
    if (threadIdx.x < 32) {                   // one wave issues the TDM descriptor
        unsigned long long ga = (unsigned long long)(const void*)(W + col0);
        unsigned lds = (unsigned)(unsigned long long)(const void*)Bs; // flat LDS addr: low 32 bits = LDS byte offset
        u32x4 g0;
        g0[0] = 1u;                                         // count=1, user mode, no gather
        g0[1] = lds;                                        // lds_addr (bytes)
        g0[2] = (unsigned)ga;                               // global_addr[31:0]
        g0[3] = (unsigned)(ga >> 32) | 0x80000000u;         // global_addr[56:32] | type=2
        i32x8 g1;
        g1[0] = 0x20000;                                    // data_size=4B, mask=0, flags=0
        g1[1] = (int)(16u << 16);                           // tensor_dim0 = 16 (tile-local)
        g1[2] = (int)(((unsigned)K & 0xFFFFu) << 16);       // td0 hi=0 | tensor_dim1 lo = K
        g1[3] = (int)(((unsigned)K >> 16) | (16u << 16));   // td1 hi | tile_dim0 = 16
        g1[4] = K & 0xFFFF;                                 // tile_dim1 = K, tile_dim2 = 0
        g1[5] = Nc;                                         // tensor_dim0_stride[31:0] = Nc
        g1[6] = 0;                                          // stride0 hi / stride1 lo
        g1[7] = 0;
        i32x4 z4 = {0, 0, 0, 0};                            // groups 2/3 unused (2D tensor)
        i32x8 z8 = {0, 0, 0, 0, 0, 0, 0, 0};
        __builtin_amdgcn_tensor_load_to_lds(g0, g1, z4, z4, z8, 0);
        __builtin_amdgcn_s_wait_tensorcnt(0);
    }
    __syncthreads();
#else
    for (int t = threadIdx.x; t < K * 16; t += 256) {
        const int k = t >> 4, n = t & 15;
        Bs[t] = W[(long long)k * Nc + col0 + n];
    }
    __syncthreads();
#endif

    const int rowTile = blockIdx.x * 8 + wave;
    const int row0 = rowTile * 16;
    if (row0 >= M) return;                    // uniform per-wave exit (EXEC all-ones for WMMA)
    const int half = lane >> 4;               // 0: lanes 0-15, 1: lanes 16-31
    const int lm   = lane & 15;

    const float* __restrict__ Arow = A + (long long)(row0 + lm) * K;
    __builtin_prefetch(Arow, 0, 3);           // global_prefetch_b8: start the A row stream

    v8f c = {};
    for (int k = 0; k < K; k += 4) {
        if ((k & 31) == 0) __builtin_prefetch(Arow + k + 32, 0, 3);
        const int kk = k + half * 2;
        v2f a; a.x = Arow[kk];            a.y = Arow[kk + 1];
        v2f b; b.x = Bs[kk * 16 + lm];    b.y = Bs[(kk + 1) * 16 + lm];
        c = __builtin_amdgcn_wmma_f32_16x16x4_f32(false, a, false, b, (short)0, c,
                                                  false, false);
    }

    const float bv = bias ? bias[col0 + lm] : 0.0f;
    float* __restrict__ Crow = C + (long long)row0 * Nc + col0 + lm;
#pragma unroll
    for (int j = 0; j < 8; ++j) {
        const int m = j + half * 8;
        Crow[(long long)m * Nc] = c[j] + bv;
    }
}

// ---------------- normalized scatter-add over edges (float4 per thread) ----------------
// agg[dst[e], f..f+3] += t[src[e], f..f+3] * dinv[src[e]] * dinv[dst[e]]
template <int F>
__global__ __launch_bounds__(256)
void scatter_kernel(const int* __restrict__ src, const int* __restrict__ dst,
                    const float* __restrict__ dinv, const float* __restrict__ t,
                    float* __restrict__ agg, long long total4) {
    long long idx = (long long)blockIdx.x * blockDim.x + threadIdx.x;
    if (idx >= total4) return;
    constexpr int Q = F / 4;                        // float4 chunks per edge
    const int e  = (int)(idx >> __builtin_ctz(Q));  // Q is power of two
    const int f4 = (int)(idx & (Q - 1)) * 4;
    const int s = src[e], d = dst[e];               // uniform within a wave for F=128
    const float norm = dinv[s] * dinv[d];
    const float4 v = *(const float4*)(t + (long long)s * F + f4);
    float* __restrict__ a = agg + (long long)d * F + f4;
    atomicAdd(a + 0, v.x * norm);
    atomicAdd(a + 1, v.y * norm);
    atomicAdd(a + 2, v.z * norm);
    atomicAdd(a + 3, v.w * norm);
}

// ---------------- combine: self-loop term + bias (+ReLU), in place ----------------
template <int F, bool RELU>
__global__ __launch_bounds__(256)
void combine_kernel(float* __restrict__ agg, const float* __restrict__ t,
                    const float* __restrict__ dinv, const float* __restrict__ bias,
                    long long total) {
    long long idx = (long long)blockIdx.x * blockDim.x + threadIdx.x;
    if (idx >= total) return;
    const int i = (int)(idx >> __builtin_ctz(F));
    const int f = (int)(idx & (F - 1));
    const float di = dinv[i];
    float v = agg[idx] + t[idx] * di * di + bias[f];
    if (RELU) v = fmaxf(v, 0.0f);
    agg[idx] = v;
}

// ---------------- per-edge bilinear score: out[e] = dot(tsrc[s], h[d]) + bb ----------------
// One wave (32 lanes) per edge; each lane covers features {lane, lane+32}; wave reduce.
__global__ __launch_bounds__(256)
void edge_score_kernel(const int* __restrict__ src, const int* __restrict__ dst,
                       const float* __restrict__ tsrc, const float* __restrict__ h,
                       const float* __restrict__ bb, float* __restrict__ out, int E) {
    const int wave = threadIdx.x >> 5;
    const int lane = threadIdx.x & 31;
    const int e = blockIdx.x * 8 + wave;
    if (e >= E) return;
    const int s = src[e], d = dst[e];
    const float* __restrict__ a = tsrc + (long long)s * F_OUT;
    const float* __restrict__ b = h    + (long long)d * F_OUT;
    float p = a[lane] * b[lane] + a[lane + 32] * b[lane + 32];
#pragma unroll
    for (int off = 16; off > 0; off >>= 1) p += __shfl_down(p, off, 32);
    if (lane == 0) out[e] = p + bb[0];
}

// ---------------- launcher ----------------

extern "C" void kernel_launch(void* const* d_in, const int* in_sizes, int n_in,
                              void* d_out, int out_size, void* d_ws, size_t ws_size,
                              hipStream_t stream) {
    (void)in_sizes; (void)n_in; (void)out_size; (void)ws_size;

    const float* x  = (const float*)d_in[0];
    const int*   ei = (const int*)d_in[1];
    const float* W0 = (const float*)d_in[2];
    const float* b0 = (const float*)d_in[3];
    const float* W1 = (const float*)d_in[4];
    const float* b1 = (const float*)d_in[5];
    const float* W2 = (const float*)d_in[6];
    const float* b2 = (const float*)d_in[7];
    const float* Wb = (const float*)d_in[8];   // [64,64] row-major
    const float* bb = (const float*)d_in[9];
    float* out = (float*)d_out;

    const int* src = ei;
    const int* dst = ei + N_EDGES;

    // workspace layout (floats); stage-2 buffers reuse the h0 region (256N >= 3*64N)
    float* ws   = (float*)d_ws;
    float* dinv = ws;                                   // N
    float* h0   = ws + N_NODES;                         // N*256
    float* t1   = h0 + (long long)N_NODES * F_IN;       // N*128
    float* h1   = t1 + (long long)N_NODES * F_HID;      // N*128 (agg1 -> h1 in place)
    float* t2   = h0;                                   // N*64  (reuse)
    float* h2   = h0 + (long long)N_NODES * F_OUT;      // N*64  (agg2 -> h2 in place)
    float* tsrc = h0 + (long long)N_NODES * (2 * F_OUT);// N*64

    const int rowTiles = N_NODES / 16;                  // 6250
    const int gemmGx   = (rowTiles + 7) / 8;            // 782 blocks of 8 waves

    // --- degrees: deg = 1 (self loop) + in-degree; dinv = deg^-1/2
    fill_kernel<<<(N_NODES + 255) / 256, 256, 0, stream>>>(dinv, 1.0f, N_NODES);
    deg_accum_kernel<<<(N_EDGES + 255) / 256, 256, 0, stream>>>(dst, dinv, N_EDGES);
    dinv_kernel<<<(N_NODES + 255) / 256, 256, 0, stream>>>(dinv, N_NODES);

    // --- h0 = x @ W0 + b0
    gemm_wmma_f32<<<dim3(gemmGx, F_IN / 16), 256, 0, stream>>>(x, W0, b0, h0,
                                                               N_NODES, F_IN, F_IN);

    // --- conv1: t1 = h0 @ W1; aggregate; h1 = relu(agg + self + b1)
    gemm_wmma_f32<<<dim3(gemmGx, F_HID / 16), 256, 0, stream>>>(h0, W1, nullptr, t1,
                                                                N_NODES, F_IN, F_HID);
    {
        long long nTot = (long long)N_NODES * F_HID;
        long long eTot4 = (long long)N_EDGES * (F_HID / 4);
        fill_kernel<<<(unsigned)((nTot + 255) / 256), 256, 0, stream>>>(h1, 0.0f, nTot);
        scatter_kernel<F_HID><<<(unsigned)((eTot4 + 255) / 256), 256, 0, stream>>>(
            src, dst, dinv, t1, h1, eTot4);
        combine_kernel<F_HID, true><<<(unsigned)((nTot + 255) / 256), 256, 0, stream>>>(
            h1, t1, dinv, b1, nTot);
    }

    // --- conv2: t2 = h1 @ W2; aggregate; h2 = agg + self + b2
    gemm_wmma_f32<<<dim3(gemmGx, F_OUT / 16), 256, 0, stream>>>(h1, W2, nullptr, t2,
                                                                N_NODES, F_HID, F_OUT);
    {
        long long nTot = (long long)N_NODES * F_OUT;
        long long eTot4 = (long long)N_EDGES * (F_OUT / 4);
        fill_kernel<<<(unsigned)((nTot + 255) / 256), 256, 0, stream>>>(h2, 0.0f, nTot);
        scatter_kernel<F_OUT><<<(unsigned)((eTot4 + 255) / 256), 256, 0, stream>>>(
            src, dst, dinv, t2, h2, eTot4);
        combine_kernel<F_OUT, false><<<(unsigned)((nTot + 255) / 256), 256, 0, stream>>>(
            h2, t2, dinv, b2, nTot);
    }

    // --- bilinear scoring: tsrc = h2 @ Wb (per node), then per-edge dot
    gemm_wmma_f32<<<dim3(gemmGx, F_OUT / 16), 256, 0, stream>>>(h2, Wb, nullptr, tsrc,
                                                                N_NODES, F_OUT, F_OUT);
    edge_score_kernel<<<(N_EDGES + 7) / 8, 256, 0, stream>>>(src, dst, tsrc, h2, bb,
                                                             out, N_EDGES);
}